// DendriticLayer_71339406787431
// MI455X (gfx1250) — compile-verified
//
#include <hip/hip_runtime.h>
#include <stdint.h>

// Fused 4-layer "dendritic" network for B=8192, I=8192, BRANCH=2, DEPTH=4.
// out[n,j] = 15-node binary tree over x[n,16j..16j+15] with leaky-ReLU at
// every node. Pure HBM-streaming workload (~274 MB -> ~12 us at 23.3 TB/s).
// CDNA5 path: global_load_async_to_lds_b128 double-buffering (ASYNCcnt),
// non-temporal hints so the 256 MB x stream doesn't sweep the 192 MB L2
// that broadcasts the (tiny) live weight band to all 1024 blocks.

#define ROWS_PER_BLOCK 16
#define NEG_SLOPE 0.1f

// One 16-byte async copy, lane-addressed, non-temporal (read-once stream).
// INST_OFFSET applies to both the global source and the LDS destination
// (CDNA5 ISA).
#define ASYNC16(LDSOFF, GADDR, OFFSTR)                                    \
  asm volatile("global_load_async_to_lds_b128 %0, %1, off offset:" OFFSTR \
               " th:TH_LOAD_NT"                                           \
               :: "v"(LDSOFF), "v"(GADDR) : "memory")

__device__ __forceinline__ void issue_row_copy(unsigned lds_byte, uint64_t gaddr) {
  ASYNC16(lds_byte, gaddr, "0");
  ASYNC16(lds_byte, gaddr, "16");
  ASYNC16(lds_byte, gaddr, "32");
  ASYNC16(lds_byte, gaddr, "48");
}

__device__ __forceinline__ float lrelu(float t) {
  return t > 0.f ? t : NEG_SLOPE * t;
}

__global__ __launch_bounds__(256) void dendritic4_fused(
    const float* __restrict__ x,
    const float* __restrict__ w0, const float* __restrict__ b0,
    const float* __restrict__ w1, const float* __restrict__ b1,
    const float* __restrict__ w2, const float* __restrict__ b2,
    const float* __restrict__ w3, const float* __restrict__ b3,
    float* __restrict__ out)
{
  // Double-buffered x slab: 2 x 256 threads x 16 floats = 32 KB LDS.
  __shared__ float4 xbuf[2][1024];

  const int jl = threadIdx.x;           // local output column 0..255
  const int jt = blockIdx.x & 1;        // which half of the 512 outputs
  const int rg = blockIdx.x >> 1;       // row group (16 rows each)
  const int j  = (jt << 8) + jl;        // global output column 0..511

  // ---- Gather the live diagonal-band weights into registers (reused 16x).
  // Only w[o, 2o..2o+1] of each (num_out x num_in) matrix is ever used:
  // flat index = o*(num_in+2), contiguous float2, 8-byte aligned.
  float2 u0[8]; float c0[8];
#pragma unroll
  for (int r = 0; r < 8; ++r) {
    const int o = 8 * j + r;
    u0[r] = *(const float2*)(w0 + (size_t)o * 8194u);
    c0[r] = b0[o];
  }
  float2 u1[4]; float c1[4];
#pragma unroll
  for (int r = 0; r < 4; ++r) {
    const int o = 4 * j + r;
    u1[r] = *(const float2*)(w1 + (size_t)o * 4098u);
    c1[r] = b1[o];
  }
  float2 u2[2]; float c2[2];
#pragma unroll
  for (int r = 0; r < 2; ++r) {
    const int o = 2 * j + r;
    u2[r] = *(const float2*)(w2 + (size_t)o * 2050u);
    c2[r] = b2[o];
  }
  const float2 u3 = *(const float2*)(w3 + (size_t)j * 1026u);
  const float  c3 = b3[j];

  // ---- Async double-buffered stream over 16 rows ----
  const int n0 = rg * ROWS_PER_BLOCK;
  const float* xbase = x + (size_t)n0 * 8192u + (size_t)jt * 4096u + (size_t)jl * 16u;
  const uint64_t g0   = (uint64_t)(uintptr_t)xbase;            // row stride 32768 B
  const unsigned lds0 = (unsigned)(uintptr_t)(&xbuf[0][jl << 2]);

  issue_row_copy(lds0, g0);  // prologue: row 0 -> buffer 0

  float* orow = out + (size_t)n0 * 512u + (size_t)j;

  for (int i = 0; i < ROWS_PER_BLOCK; ++i) {
    if (i + 1 < ROWS_PER_BLOCK) {
      // Prefetch next row into the other buffer, then wait for the 4 oldest
      // copies (current buffer). Async loads retire in order per wave.
      issue_row_copy(lds0 + (unsigned)(((i + 1) & 1) << 14),
                     g0 + (uint64_t)(i + 1) * 32768u);
      asm volatile("s_wait_asynccnt 0x4" ::: "memory");
    } else {
      asm volatile("s_wait_asynccnt 0x0" ::: "memory");
    }

    const float4* xb = &xbuf[i & 1][jl << 2];
    const float4 q0 = xb[0], q1 = xb[1], q2 = xb[2], q3 = xb[3];
    const float v[16] = {q0.x, q0.y, q0.z, q0.w,  q1.x, q1.y, q1.z, q1.w,
                         q2.x, q2.y, q2.z, q2.w,  q3.x, q3.y, q3.z, q3.w};

    float a0[8];
#pragma unroll
    for (int r = 0; r < 8; ++r)
      a0[r] = lrelu(fmaf(v[2 * r], u0[r].x, fmaf(v[2 * r + 1], u0[r].y, c0[r])));

    float a1[4];
#pragma unroll
    for (int r = 0; r < 4; ++r)
      a1[r] = lrelu(fmaf(a0[2 * r], u1[r].x, fmaf(a0[2 * r + 1], u1[r].y, c1[r])));

    float a2[2];
#pragma unroll
    for (int r = 0; r < 2; ++r)
      a2[r] = lrelu(fmaf(a1[2 * r], u2[r].x, fmaf(a1[2 * r + 1], u2[r].y, c2[r])));

    const float res = lrelu(fmaf(a2[0], u3.x, fmaf(a2[1], u3.y, c3)));
    __builtin_nontemporal_store(res, &orow[(size_t)i * 512u]);
  }
}

extern "C" void kernel_launch(void* const* d_in, const int* in_sizes, int n_in,
                              void* d_out, int out_size, void* d_ws, size_t ws_size,
                              hipStream_t stream) {
  (void)in_sizes; (void)n_in; (void)out_size; (void)d_ws; (void)ws_size;
  const float* x  = (const float*)d_in[0];
  const float* w0 = (const float*)d_in[1];
  const float* b0 = (const float*)d_in[2];
  const float* w1 = (const float*)d_in[3];
  const float* b1 = (const float*)d_in[4];
  const float* w2 = (const float*)d_in[5];
  const float* b2 = (const float*)d_in[6];
  const float* w3 = (const float*)d_in[7];
  const float* b3 = (const float*)d_in[8];
  float* out = (float*)d_out;

  // 2 j-tiles (256 outputs each) x 512 row-groups (16 rows each) = 1024 blocks.
  const dim3 grid(2 * (8192 / ROWS_PER_BLOCK));
  const dim3 block(256);
  dendritic4_fused<<<grid, block, 0, stream>>>(x, w0, b0, w1, b1, w2, b2, w3, b3, out);
}